// Network_52295521796159
// MI455X (gfx1250) — compile-verified
//
#include <hip/hip_runtime.h>
#include <math.h>

#define POOL 7
#define PRE 14
#define NSAMP (PRE * PRE)      /* 196 */
#define C_CH 256
#define OUT_HW (POOL * POOL)   /* 49  */

typedef __attribute__((ext_vector_type(4))) unsigned int u32x4;
typedef __attribute__((ext_vector_type(8))) unsigned int u32x8;

/* ---- CDNA5 async global<->LDS helpers (ASYNCcnt path, gfx1250) ----------- */
__device__ __forceinline__ unsigned lds_off_u32(const void* p)
{
    /* ISA 10.2 aperture rule: LDS address is the low 32 bits of the pointer */
    return (unsigned)(uintptr_t)p;
}
__device__ __forceinline__ void async_store_b32(void* gaddr, unsigned lds_addr)
{
    asm volatile("global_store_async_from_lds_b32 %0, %1, off"
                 :: "v"(gaddr), "v"(lds_addr) : "memory");
}

/* ---- CDNA5 Tensor Data Mover (TENSORcnt path, gfx1250) ------------------- */
/* D# group0 (CDNA5 ISA): count=1, lds_addr, 57b global, type=2 */
__device__ __forceinline__ u32x4 tdm_group0(unsigned lds_addr, const void* gptr)
{
    unsigned long long ga = (unsigned long long)(uintptr_t)gptr;
    u32x4 g0;
    g0.x = 1u;                                           /* count=1 (valid)   */
    g0.y = lds_addr;                                     /* LDS byte address  */
    g0.z = (unsigned)ga;                                 /* global_addr[31:0] */
    g0.w = (unsigned)((ga >> 32) & 0x1FFFFFFu) | (2u << 30); /* [56:32]|type=2 */
    return g0;
}
/* D# group1 (8.4): data_size=4B, dims/strides in elements, optional LDS pad  */
__device__ __forceinline__ u32x8 tdm_group1(unsigned dim0, unsigned dim1,
                                            unsigned til0, unsigned til1,
                                            unsigned long long str0,
                                            unsigned pad_en, unsigned pad_ivl,
                                            unsigned pad_amt)
{
    u32x8 g1;
    g1.s0 = (2u << 16) | (pad_en << 20) | (pad_ivl << 22) | (pad_amt << 25);
    g1.s1 = (dim0 & 0xFFFFu) << 16;                      /* tensor_dim0[15:0] */
    g1.s2 = (dim0 >> 16) | ((dim1 & 0xFFFFu) << 16);     /* dim0[31:16]|dim1  */
    g1.s3 = (dim1 >> 16) | (til0 << 16);                 /* dim1[31:16]|tile0 */
    g1.s4 = til1;                                        /* tile1 | tile2=0   */
    g1.s5 = (unsigned)str0;                              /* stride0[31:0]     */
    g1.s6 = (unsigned)(str0 >> 32);                      /* stride0[47:32]    */
    g1.s7 = 0u;                                          /* stride1 unused    */
    return g1;
}
__device__ __forceinline__ void tdm_load(u32x4 g0, u32x8 g1)
{
    asm volatile("tensor_load_to_lds %0, %1" :: "s"(g0), "s"(g1) : "memory");
}
__device__ __forceinline__ void tdm_store(u32x4 g0, u32x8 g1)
{
    asm volatile("tensor_store_from_lds %0, %1" :: "s"(g0), "s"(g1) : "memory");
}

/* ---------------- per-ROI params (matches reference exactly) -------------- */
__device__ __forceinline__ void roi_params(const float* __restrict__ rois, int n,
                                           int& lvl, int& HW,
                                           float& x1s, float& y1s,
                                           float& x2s, float& y2s)
{
    float x1 = rois[n * 5 + 1], y1 = rois[n * 5 + 2];
    float x2 = rois[n * 5 + 3], y2 = rois[n * 5 + 4];
    float w = x2 - x1; if (w <= 0.0f) w = 1e-14f;
    float h = y2 - y1; if (h <= 0.0f) h = 1e-14f;
    float kf = 4.0f + log2f(sqrtf(w * h) * (1.0f / 224.0f));
    kf = fminf(fmaxf(kf, 2.0f), 5.0f);
    float k = rintf(kf);              /* round-half-even == jnp.round */
    lvl = (int)k - 2;                 /* 0..3 */
    HW  = 256 >> lvl;
    float inv = exp2f(-k);            /* exact power of two: x/scale == x*inv */
    x1s = x1 * inv; y1s = y1 * inv; x2s = x2 * inv; y2s = y2 * inv;
}

/* --------- one 14x14 sample record: 4 tap offsets + masked weights -------- */
__device__ __forceinline__ void sample_record(int s, float x1s, float y1s,
                                              float x2s, float y2s, int HW,
                                              int* off, float* wgt)
{
    int sy = s / PRE, sx = s % PRE;
    const float step = 2.0f / (float)(PRE - 1);
    float tx = -1.0f + (float)sx * step;
    float ty = -1.0f + (float)sy * step;
    float mx = (x1s + x2s) * 0.5f, hx = (x2s - x1s) * 0.5f;
    float my = (y1s + y2s) * 0.5f, hy = (y2s - y1s) * 0.5f;
    float px = mx + tx * hx;
    float py = my + ty * hy;
    float x0 = floorf(px), y0 = floorf(py);
    float dx = px - x0,    dy = py - y0;
    int xi = (int)x0, yi = (int)y0;
    float w00 = (1.0f - dy) * (1.0f - dx);
    float w01 = (1.0f - dy) * dx;
    float w10 = dy * (1.0f - dx);
    float w11 = dy * dx;
    int   xs[4] = { xi, xi + 1, xi,     xi + 1 };
    int   ys[4] = { yi, yi,     yi + 1, yi + 1 };
    float wv[4] = { w00, w01, w10, w11 };
#pragma unroll
    for (int t = 0; t < 4; ++t) {
        bool inb = (xs[t] >= 0) && (xs[t] < HW) && (ys[t] >= 0) && (ys[t] < HW);
        int xc = min(max(xs[t], 0), HW - 1);
        int yc = min(max(ys[t], 0), HW - 1);
        off[t] = yc * HW + xc;
        wgt[t] = inb ? wv[t] : 0.0f;
    }
}

/* ------------- pass 1: (C,H,W) -> (H,W,C) into workspace ------------------ */
/* TDM loads the whole 256(ch) x 32(spatial) strided tile into LDS in ONE     */
/* instruction, with pad_enable giving LDS row stride 33 floats (odd stride   */
/* -> conflict-free transposed column reads). Store phase streams channel-    */
/* contiguous LDS rows straight to global with async-from-LDS DMA.            */
__global__ __launch_bounds__(256) void transpose_chw_hwc(
    const float* __restrict__ p2, const float* __restrict__ p3,
    const float* __restrict__ p4, const float* __restrict__ p5,
    float* __restrict__ ws)
{
    int b = blockIdx.x;
    const float* src; float* dst; int spat; int tile;
    if (b < 2048)      { src = p2; dst = ws;                                  spat = 65536; tile = b;        }
    else if (b < 2560) { src = p3; dst = ws + (size_t)65536 * 256;            spat = 16384; tile = b - 2048; }
    else if (b < 2688) { src = p4; dst = ws + (size_t)(65536 + 16384) * 256;  spat = 4096;  tile = b - 2560; }
    else               { src = p5; dst = ws + (size_t)(65536 + 16384 + 4096) * 256; spat = 1024; tile = b - 2688; }

    __shared__ float t2[256][33];        /* 33792 B; stride 33 via TDM pad    */
    int tid = threadIdx.x;
    int s0  = tile * 32;
    unsigned tbase = lds_off_u32(&t2[0][0]);

    if (tid < 32) {                      /* wave 0 issues the tensor DMA      */
        /* 2D tile: rows = 256 channels (stride spat), row = 32 floats.      */
        /* pad: every 32 DWORDs (pad_ivl enc 4) insert 1 DWORD (enc 0).      */
        u32x4 g0 = tdm_group0(tbase, src + s0);
        u32x8 g1 = tdm_group1(/*dim0*/(unsigned)spat, /*dim1*/256u,
                              /*til0*/32u, /*til1*/256u,
                              /*str0*/(unsigned long long)(unsigned)spat,
                              /*pad*/1u, /*ivl*/4u, /*amt*/0u);
        tdm_load(g0, g1);
        __builtin_amdgcn_s_wait_tensorcnt(0);
    }
    __syncthreads();

    /* t2[c][s] -> dst[(s0+s)*256 + c]: async DMA LDS->global, 1KB/iteration */
    for (int s = 0; s < 32; ++s)
        async_store_b32(dst + (size_t)(s0 + s) * 256 + tid,
                        tbase + (unsigned)(tid * 33 + s) * 4u);
    /* s_endpgm drains ASYNCcnt */
}

/* ------------- pass 2: ROI-align + maxpool on (H,W,C) layout ------------- */
__global__ __launch_bounds__(256) void roi_pool_trans(
    const float* __restrict__ ws, const float* __restrict__ rois,
    float* __restrict__ out)
{
    __shared__ int   s_off[NSAMP * 4];        /* spatial offset * 256         */
    __shared__ float s_wgt[NSAMP * 4];
    __shared__ __align__(16) float s_out[C_CH * OUT_HW]; /* 50176 B staging   */

    int n = blockIdx.x, tid = threadIdx.x;
    int lvl, HW; float x1s, y1s, x2s, y2s;
    roi_params(rois, n, lvl, HW, x1s, y1s, x2s, y2s);

    if (tid < NSAMP) {
        int off[4]; float wgt[4];
        sample_record(tid, x1s, y1s, x2s, y2s, HW, off, wgt);
#pragma unroll
        for (int t = 0; t < 4; ++t) {
            s_off[tid * 4 + t] = off[t] * C_CH;
            s_wgt[tid * 4 + t] = wgt[t];
        }
    }
    __syncthreads();

    const size_t lvl_off[4] = { 0, (size_t)65536 * 256,
                                (size_t)(65536 + 16384) * 256,
                                (size_t)(65536 + 16384 + 4096) * 256 };
    const float* base = ws + lvl_off[lvl];

    int q = tid & 63;     /* channel quad: channels 4q..4q+3 (consec lanes -> 512B coalesced) */
    int g = tid >> 6;     /* output-pixel group */

    for (int o = g; o < OUT_HW; o += 4) {
        int oy = o / POOL, ox = o % POOL;
        float4 vq[4];
#pragma unroll
        for (int a = 0; a < 2; ++a)
#pragma unroll
            for (int bb = 0; bb < 2; ++bb) {
                int s = (2 * oy + a) * PRE + (2 * ox + bb);
                float4 v = make_float4(0.f, 0.f, 0.f, 0.f);
#pragma unroll
                for (int t = 0; t < 4; ++t) {
                    const float4 gv = *(const float4*)(base + s_off[s * 4 + t] + q * 4);
                    float w = s_wgt[s * 4 + t];
                    v.x = fmaf(w, gv.x, v.x); v.y = fmaf(w, gv.y, v.y);
                    v.z = fmaf(w, gv.z, v.z); v.w = fmaf(w, gv.w, v.w);
                }
                vq[a * 2 + bb] = v;
            }
        float4 m;
        m.x = fmaxf(fmaxf(vq[0].x, vq[1].x), fmaxf(vq[2].x, vq[3].x));
        m.y = fmaxf(fmaxf(vq[0].y, vq[1].y), fmaxf(vq[2].y, vq[3].y));
        m.z = fmaxf(fmaxf(vq[0].z, vq[1].z), fmaxf(vq[2].z, vq[3].z));
        m.w = fmaxf(fmaxf(vq[0].w, vq[1].w), fmaxf(vq[2].w, vq[3].w));
        s_out[(4 * q + 0) * OUT_HW + o] = m.x;
        s_out[(4 * q + 1) * OUT_HW + o] = m.y;
        s_out[(4 * q + 2) * OUT_HW + o] = m.z;
        s_out[(4 * q + 3) * OUT_HW + o] = m.w;
    }
    __syncthreads();

    /* writeback: one 1D tensor_store_from_lds moves the whole 50KB tile      */
    if (tid < 32) {
        float* outn = out + (size_t)n * (C_CH * OUT_HW);
        u32x4 g0 = tdm_group0(lds_off_u32(s_out), outn);
        u32x8 g1 = tdm_group1(/*dim0*/(unsigned)(C_CH * OUT_HW), /*dim1*/1u,
                              /*til0*/(unsigned)(C_CH * OUT_HW), /*til1*/1u,
                              /*str0*/(unsigned long long)(C_CH * OUT_HW),
                              /*pad*/0u, /*ivl*/0u, /*amt*/0u);
        tdm_store(g0, g1);
        __builtin_amdgcn_s_wait_tensorcnt(0);
    }
}

/* --------- fallback: direct gathers on native (C,H,W) layout -------------- */
__global__ __launch_bounds__(256) void roi_pool_direct(
    const float* __restrict__ p2, const float* __restrict__ p3,
    const float* __restrict__ p4, const float* __restrict__ p5,
    const float* __restrict__ rois, float* __restrict__ out)
{
    __shared__ int   s_off[NSAMP * 4];
    __shared__ float s_wgt[NSAMP * 4];

    int n = blockIdx.x, tid = threadIdx.x;
    int lvl, HW; float x1s, y1s, x2s, y2s;
    roi_params(rois, n, lvl, HW, x1s, y1s, x2s, y2s);

    if (tid < NSAMP) {
        int off[4]; float wgt[4];
        sample_record(tid, x1s, y1s, x2s, y2s, HW, off, wgt);
#pragma unroll
        for (int t = 0; t < 4; ++t) {
            s_off[tid * 4 + t] = off[t];
            s_wgt[tid * 4 + t] = wgt[t];
        }
    }
    __syncthreads();

    const float* lp = (lvl == 0) ? p2 : (lvl == 1) ? p3 : (lvl == 2) ? p4 : p5;
    int plane = HW * HW;
    int wv = tid >> 5, lane = tid & 31;
    float* outn = out + (size_t)n * (C_CH * OUT_HW);

    for (int o = lane; o < OUT_HW; o += 32) {
        int oy = o / POOL, ox = o % POOL;
        int so[16]; float sw[16];
#pragma unroll
        for (int a = 0; a < 2; ++a)
#pragma unroll
            for (int bb = 0; bb < 2; ++bb) {
                int s = (2 * oy + a) * PRE + (2 * ox + bb);
#pragma unroll
                for (int t = 0; t < 4; ++t) {
                    so[(a * 2 + bb) * 4 + t] = s_off[s * 4 + t];
                    sw[(a * 2 + bb) * 4 + t] = s_wgt[s * 4 + t];
                }
            }
        for (int c = wv; c < C_CH; c += 8) {
            const float* pc = lp + (size_t)c * plane;
            if (c + 8 < C_CH)
                __builtin_prefetch(lp + (size_t)(c + 8) * plane + so[0], 0, 0);
            float v[4];
#pragma unroll
            for (int qq = 0; qq < 4; ++qq) {
                float acc = 0.0f;
#pragma unroll
                for (int t = 0; t < 4; ++t)
                    acc = fmaf(sw[qq * 4 + t], pc[so[qq * 4 + t]], acc);
                v[qq] = acc;
            }
            outn[(size_t)c * OUT_HW + o] =
                fmaxf(fmaxf(v[0], v[1]), fmaxf(v[2], v[3]));
        }
    }
}

extern "C" void kernel_launch(void* const* d_in, const int* in_sizes, int n_in,
                              void* d_out, int out_size, void* d_ws, size_t ws_size,
                              hipStream_t stream)
{
    const float* p2   = (const float*)d_in[0];
    const float* p3   = (const float*)d_in[1];
    const float* p4   = (const float*)d_in[2];
    const float* p5   = (const float*)d_in[3];
    const float* rois = (const float*)d_in[4];
    float* out = (float*)d_out;
    int N = in_sizes[4] / 5;

    const size_t need = (size_t)(65536 + 16384 + 4096 + 1024) * 256 * sizeof(float); /* 89.1 MB */
    if (d_ws && ws_size >= need) {
        transpose_chw_hwc<<<2720, 256, 0, stream>>>(p2, p3, p4, p5, (float*)d_ws);
        roi_pool_trans<<<N, 256, 0, stream>>>((const float*)d_ws, rois, out);
    } else {
        roi_pool_direct<<<N, 256, 0, stream>>>(p2, p3, p4, p5, rois, out);
    }
}